// PseudoDDSPVocoder_17257178596000
// MI455X (gfx1250) — compile-verified
//
#include <hip/hip_runtime.h>
#include <cstdint>

// ---------------- problem constants ----------------
#define NB    16
#define LFR   2000
#define HOP_  240
#define IRL   512
#define NFFT  480           // 2*HOP
#define CAPG  8192          // grain capacity per batch (max real ~6000)
#define SR_F  24000.0f
#define PI_F  3.14159265358979323846f

#define SIGLEN   (LFR*HOP_ + IRL)       // 480512
#define NOISELEN ((LFR+1)*HOP_)         // 480240
#define OUTLEN   (LFR*HOP_)             // 480000

// workspace layout (32-bit words)
#define SIG_W    ((size_t)NB*SIGLEN)                  // 7,688,192 floats
#define NOISE_W  ((size_t)NB*NOISELEN)                // 7,683,840 floats
#define CNT_OFF  (SIG_W + NOISE_W)                    // 16 ints
#define GT_OFF   (CNT_OFF + 16)                       // 16*CAPG ints
#define GF_OFF   (GT_OFF + (size_t)NB*CAPG)           // 16*CAPG floats
#define ZERO_W   (CNT_OFF + 16)                       // words to zero each call

typedef float v2f  __attribute__((ext_vector_type(2)));
typedef float v8f  __attribute__((ext_vector_type(8)));
typedef unsigned int u32x4 __attribute__((ext_vector_type(4)));
typedef int   i32x4 __attribute__((ext_vector_type(4)));
typedef int   i32x8 __attribute__((ext_vector_type(8)));

__device__ __forceinline__ v8f wmma4(v2f a, v2f b, v8f c) {
  // D = A(16x4) * B(4x16) + C(16x16), fp32
  return __builtin_amdgcn_wmma_f32_16x16x4_f32(false, a, false, b, (short)0, c, false, false);
}

// ---------------- zero scratch ----------------
__global__ void k_zero(unsigned int* __restrict__ w, long n) {
  long i = (long)blockIdx.x * blockDim.x + threadIdx.x;
  long s = (long)gridDim.x * blockDim.x;
  for (; i < n; i += s) w[i] = 0u;
}

// ---------------- phase accumulation + wrap (grain) detection ----------------
// one block per batch
__global__ void k_phase(const float* __restrict__ pitch, const float* __restrict__ phase0,
                        int* __restrict__ counters, int* __restrict__ gT, float* __restrict__ gF) {
  __shared__ float baseC[LFR];   // phase at last sample of previous frame (mod 1)
  const int b = blockIdx.x;
  const float* pb = pitch + (size_t)b * LFR;
  const float p0 = phase0[b];
  if (threadIdx.x == 0) {
    // frame 0: nf[0]=phase0, nf[1..239]=step0  -> P[239] = phase0 + 239*step0 (mod 1)
    float c = p0 + 239.f * pb[0] * (1.f / SR_F);
    c -= floorf(c);
    baseC[0] = 0.f;
    if (LFR > 1) baseC[1] = c;
    for (int l = 1; l < LFR - 1; ++l) {
      c = c + 240.f * pb[l] * (1.f / SR_F);
      c -= floorf(c);
      baseC[l + 1] = c;
    }
  }
  __syncthreads();
  const int T = LFR * HOP_;
  for (int t = threadIdx.x; t < T - 1; t += blockDim.x) {
    int l0 = t / HOP_, u0 = t - l0 * HOP_;
    float s0 = pb[l0] * (1.f / SR_F);
    float pA = (l0 == 0) ? (p0 + (float)t * s0) : (baseC[l0] + (float)(u0 + 1) * s0);
    pA -= floorf(pA);
    int t1 = t + 1;
    int l1 = t1 / HOP_, u1 = t1 - l1 * HOP_;
    float s1 = pb[l1] * (1.f / SR_F);
    float pB = (l1 == 0) ? (p0 + (float)t1 * s1) : (baseC[l1] + (float)(u1 + 1) * s1);
    pB -= floorf(pB);
    if (pA > pB) {                       // phase wrapped -> grain onset
      float numer = 1.f - pA;
      float frac = numer / (numer + pB);
      int idx = atomicAdd(&counters[b], 1);
      if (idx < CAPG) {
        gT[b * CAPG + idx] = t;
        gF[b * CAPG + idx] = frac;
      }
    }
  }
}

// ---------------- grain synthesis: DFT512 -> frac rotate -> IDFT512 -> scatter ----------------
// one block = 16 grains; 8 waves; WMMA f32 16x16x4 GEMMs against on-the-fly trig basis.
__global__ void __launch_bounds__(256) k_grain(const float* __restrict__ ir,
                                               const int* __restrict__ counters,
                                               const int* __restrict__ gT,
                                               const float* __restrict__ gF,
                                               float* __restrict__ sig) {
  __shared__ float sG[IRL * 16];    // grains, [m][j], 32 KB
  __shared__ float sR[256 * 16];    // cos sums -> scaled Re(spec), 16 KB
  __shared__ float sI[256 * 16];    // sin sums -> scaled Im(spec); row0 reused as Nyquist stash, 16 KB

  const int tilesPerB = CAPG / 16;
  const int b = blockIdx.x / tilesPerB;
  const int tile = blockIdx.x % tilesPerB;
  int count = counters[b];
  if (count > CAPG) count = CAPG;
  const int base = tile * 16;
  if (base >= count) return;                       // uniform per block

  const int tid = threadIdx.x;
  // stage 16 grain columns ir[b, :, t//HOP]
  for (int e = 0; e < 32; ++e) {
    int lin = tid + 256 * e;                       // 8192 elements
    int m = lin >> 4, j = lin & 15;
    float v = 0.f;
    int jg = base + j;
    if (jg < count) {
      int frame = gT[b * CAPG + jg] / HOP_;
      v = ir[((size_t)b * IRL + m) * LFR + frame];
    }
    sG[lin] = v;
  }
  __syncthreads();

  const int lane = tid & 31;
  const int w = tid >> 5;
  const int col = lane & 15;
  const int half = lane >> 4;                      // K-half within A/B fragments

  // ---- forward: R[k]=sum_m g[m]cos(2pi k m/512), S[k]=sum_m g[m]sin(...), k in [0,256) ----
  {
    v8f aR0 = {0,0,0,0,0,0,0,0}, aS0 = {0,0,0,0,0,0,0,0};
    v8f aR1 = {0,0,0,0,0,0,0,0}, aS1 = {0,0,0,0,0,0,0,0};
    const int k0 = (2 * w + 0) * 16 + col;
    const int k1 = (2 * w + 1) * 16 + col;
    const float th0 = (2.f * PI_F / 512.f) * (float)k0;
    const float th1 = (2.f * PI_F / 512.f) * (float)k1;
    const float c1a = __cosf(th0), s1a = __sinf(th0);
    const float c4a = __cosf(4.f * th0), s4a = __sinf(4.f * th0);
    const float c1b = __cosf(th1), s1b = __sinf(th1);
    const float c4b = __cosf(4.f * th1), s4b = __sinf(4.f * th1);
    const float m0 = (float)(2 * half);
    float ca = __cosf(th0 * m0), sa = __sinf(th0 * m0);
    float cb = __cosf(th1 * m0), sb = __sinf(th1 * m0);
    #pragma unroll 4
    for (int c = 0; c < 128; ++c) {
      int mr = 4 * c + 2 * half;
      v2f bf; bf[0] = sG[mr * 16 + col]; bf[1] = sG[(mr + 1) * 16 + col];
      v2f aC0, aS0f, aC1, aS1f;
      aC0[0] = ca; aS0f[0] = sa;
      aC0[1] = ca * c1a - sa * s1a; aS0f[1] = sa * c1a + ca * s1a;
      aC1[0] = cb; aS1f[0] = sb;
      aC1[1] = cb * c1b - sb * s1b; aS1f[1] = sb * c1b + cb * s1b;
      aR0 = wmma4(aC0, bf, aR0);
      aS0 = wmma4(aS0f, bf, aS0);
      aR1 = wmma4(aC1, bf, aR1);
      aS1 = wmma4(aS1f, bf, aS1);
      float t;
      t = ca * c4a - sa * s4a; sa = sa * c4a + ca * s4a; ca = t;
      t = cb * c4b - sb * s4b; sb = sb * c4b + cb * s4b; cb = t;
    }
    #pragma unroll
    for (int r = 0; r < 8; ++r) {
      int Mr = r + 8 * half;
      sR[((2 * w + 0) * 16 + Mr) * 16 + col] = aR0[r];
      sI[((2 * w + 0) * 16 + Mr) * 16 + col] = aS0[r];
      sR[((2 * w + 1) * 16 + Mr) * 16 + col] = aR1[r];
      sI[((2 * w + 1) * 16 + Mr) * 16 + col] = aS1[r];
    }
  }
  __syncthreads();

  // ---- rotate by frac and fold irfft weights; I = -S; spec_hat = spec * e^{-i 2pi k frac/512} ----
  for (int q = 0; q < 16; ++q) {
    int idx = tid + 256 * q;                       // 4096
    int k = idx >> 4, j = idx & 15;
    int jg = base + j;
    float frac = (jg < count) ? gF[b * CAPG + jg] : 0.f;
    float phi = (2.f * PI_F / 512.f) * (float)k * frac;
    float cp = __cosf(phi), sp = __sinf(phi);
    float R = sR[idx], S = sI[idx];
    float Rh = R * cp - S * sp;                    // Re(hat)
    float Ih = -S * cp - R * sp;                   // Im(hat)
    float wgt = ((k == 0) ? 1.f : 2.f) * (1.f / 512.f);
    sR[idx] = wgt * Rh;
    sI[idx] = wgt * Ih;
  }
  __syncthreads();
  // Nyquist bin stash in sI row 0 (its sin-basis weight in the IDFT GEMM is exactly 0)
  if (tid < 16) {
    float acc = 0.f;
    for (int m = 0; m < IRL; m += 2) acc += sG[m * 16 + tid] - sG[(m + 1) * 16 + tid];
    int jg = base + tid;
    float frac = (jg < count) ? gF[b * CAPG + jg] : 0.f;
    sI[tid] = (1.f / 512.f) * acc * __cosf(PI_F * frac);
  }
  __syncthreads();

  // ---- inverse: g'[n] = sum_k sR[k]cos(2pi k n/512) - sI[k]sin(...) + nyq*(-1)^n ----
  const float nyq = sI[col];
  #pragma unroll
  for (int tt = 0; tt < 4; ++tt) {
    int nt = 4 * w + tt;
    int nrow = nt * 16 + col;
    float th = (2.f * PI_F / 512.f) * (float)nrow;
    float c4 = __cosf(4.f * th), s4 = __sinf(4.f * th);
    float c1 = __cosf(th), s1 = __sinf(th);
    float k0f = (float)(2 * half);
    float ca = __cosf(th * k0f), sa = __sinf(th * k0f);
    v8f acc = {0,0,0,0,0,0,0,0};
    #pragma unroll 4
    for (int c = 0; c < 64; ++c) {
      int kr = 4 * c + 2 * half;
      v2f bR; bR[0] = sR[kr * 16 + col]; bR[1] = sR[(kr + 1) * 16 + col];
      v2f bI; bI[0] = sI[kr * 16 + col]; bI[1] = sI[(kr + 1) * 16 + col];
      v2f aC, aSn;
      aC[0] = ca; aSn[0] = -sa;
      float cn = ca * c1 - sa * s1, sn = sa * c1 + ca * s1;
      aC[1] = cn; aSn[1] = -sn;
      acc = wmma4(aC, bR, acc);
      acc = wmma4(aSn, bI, acc);
      float t = ca * c4 - sa * s4; sa = sa * c4 + ca * s4; ca = t;
    }
    int jg = base + col;
    if (jg < count) {
      int tpos = gT[b * CAPG + jg];
      #pragma unroll
      for (int r = 0; r < 8; ++r) {
        int n = nt * 16 + r + 8 * half;
        float val = acc[r] + ((n & 1) ? -nyq : nyq);
        atomicAdd(&sig[(size_t)b * SIGLEN + tpos + n], val);
      }
    }
  }
}

// ---------------- noise path: frame via TDM -> DFT480 -> scale -> IDFT480 -> hann -> OLA ----------------
// one block = 16 consecutive frames of one batch (2000/16 = 125 tiles per batch, never straddles b)
__global__ void __launch_bounds__(256) k_noise(const float* __restrict__ ap,
                                               const float* __restrict__ exc,
                                               float* __restrict__ noise) {
  __shared__ float smem[15360];                    // 61.4 KB, single LDS object (offset 0)
  float* sStage = smem;                            // [j][i] 16x480 (TDM dest), dead after transpose
  float* sRe = smem;                               // [row][j] 240x16 (aliases stage)
  float* sS  = smem + 3840;                        // [row][j] 240x16
  float* sX  = smem + 7680;                        // [i][j]  480x16

  const int fb = blockIdx.x * 16;
  const int b = fb / LFR;
  const int l0 = fb % LFR;
  const int tid = threadIdx.x;

  // TDM: 2D tile, 16 rows of 480 fp32, row stride 240 fp32 == the framing op
  if (tid < 32) {
    const float* gsrc = exc + (size_t)b * NOISELEN + (size_t)l0 * HOP_;
    unsigned long long ga = (unsigned long long)(uintptr_t)gsrc;
    u32x4 g0;
    g0[0] = 1u;                                            // count=1
    g0[1] = 0u;                                            // lds_addr: smem is the only LDS object -> offset 0
    g0[2] = (unsigned)(ga & 0xFFFFFFFFull);
    g0[3] = (unsigned)((ga >> 32) & 0x1FFFFFFull) | (2u << 30);  // addr hi | type=2
    i32x8 g1;
    g1[0] = (int)(2u << 16);                               // data_size=4B, mask=0
    g1[1] = (int)((NOISELEN & 0xFFFFu) << 16);             // tensor_dim0 lo16
    g1[2] = (int)((NOISELEN >> 16) | (16u << 16));         // tensor_dim0 hi16 | tensor_dim1=16
    g1[3] = (int)(480u << 16);                             // tile_dim0=480
    g1[4] = 16;                                            // tile_dim1=16, tile_dim2=0
    g1[5] = HOP_;                                          // tensor_dim0_stride=240
    g1[6] = 0; g1[7] = 0;
    i32x4 gz; gz[0] = 0; gz[1] = 0; gz[2] = 0; gz[3] = 0;
    i32x8 gz8; gz8[0] = 0; gz8[1] = 0; gz8[2] = 0; gz8[3] = 0;
    gz8[4] = 0; gz8[5] = 0; gz8[6] = 0; gz8[7] = 0;
    __builtin_amdgcn_tensor_load_to_lds(g0, g1, gz, gz, gz8, 0);
    __builtin_amdgcn_s_wait_tensorcnt(0);
  }
  __syncthreads();
  // transpose [j][i] -> [i][j] for conflict-free B fragments
  float tv[30];
  #pragma unroll
  for (int e = 0; e < 30; ++e) tv[e] = sStage[tid + 256 * e];
  __syncthreads();
  #pragma unroll
  for (int e = 0; e < 30; ++e) {
    int idx = tid + 256 * e;
    int j = idx / NFFT, i = idx - j * NFFT;
    sX[i * 16 + j] = tv[e];
  }
  __syncthreads();

  const int lane = tid & 31;
  const int w = tid >> 5;
  const int col = lane & 15;
  const int half = lane >> 4;

  // ---- forward: bins k=1..240 (rows 0..239), K = 480 samples ----
  for (int kt = w; kt < 15; kt += 8) {
    int krow = 1 + kt * 16 + col;
    float th = (2.f * PI_F / (float)NFFT) * (float)krow;
    float c1 = __cosf(th), s1 = __sinf(th);
    float c4 = __cosf(4.f * th), s4 = __sinf(4.f * th);
    float i0 = (float)(2 * half);
    float ca = __cosf(th * i0), sa = __sinf(th * i0);
    v8f aR = {0,0,0,0,0,0,0,0}, aS = {0,0,0,0,0,0,0,0};
    #pragma unroll 4
    for (int c = 0; c < 120; ++c) {
      int irow = 4 * c + 2 * half;
      v2f bf; bf[0] = sX[irow * 16 + col]; bf[1] = sX[(irow + 1) * 16 + col];
      v2f aC, aSf;
      aC[0] = ca; aSf[0] = sa;
      float cn = ca * c1 - sa * s1, sn = sa * c1 + ca * s1;
      aC[1] = cn; aSf[1] = sn;
      aR = wmma4(aC, bf, aR);
      aS = wmma4(aSf, bf, aS);
      float t = ca * c4 - sa * s4; sa = sa * c4 + ca * s4; ca = t;
    }
    #pragma unroll
    for (int r = 0; r < 8; ++r) {
      int Mr = r + 8 * half;
      sRe[(kt * 16 + Mr) * 16 + col] = aR[r];
      sS[(kt * 16 + Mr) * 16 + col]  = aS[r];
    }
  }
  __syncthreads();

  // ---- scale by aperiodicity and irfft weights (DC dropped; Nyquist weight 1/480) ----
  for (int q = 0; q < 15; ++q) {
    int idx = tid + 256 * q;                       // 3840
    int row = idx >> 4, j = idx & 15;              // k = row+1
    float sc = ap[((size_t)b * HOP_ + row) * LFR + (l0 + j)];
    float wgt = ((row == 239) ? 1.f : 2.f) * (1.f / (float)NFFT);
    float f = sc * wgt;
    sRe[idx] *= f;
    sS[idx]  *= f;
  }
  __syncthreads();

  // ---- inverse: y[n] = sum_{k=1..240} sRe*cos(2pi k n/480) + sS*sin(...); hann; OLA ----
  for (int nt = w; nt < 30; nt += 8) {
    int nrow = nt * 16 + col;
    float th = (2.f * PI_F / (float)NFFT) * (float)nrow;
    float c1 = __cosf(th), s1 = __sinf(th);
    float c4 = __cosf(4.f * th), s4 = __sinf(4.f * th);
    float a0 = th * (float)(1 + 2 * half);         // K-index r holds k=r+1
    float ca = __cosf(a0), sa = __sinf(a0);
    v8f acc = {0,0,0,0,0,0,0,0};
    #pragma unroll 4
    for (int c = 0; c < 60; ++c) {
      int kr = 4 * c + 2 * half;
      v2f bR; bR[0] = sRe[kr * 16 + col]; bR[1] = sRe[(kr + 1) * 16 + col];
      v2f bS; bS[0] = sS[kr * 16 + col];  bS[1] = sS[(kr + 1) * 16 + col];
      v2f aC, aSf;
      aC[0] = ca; aSf[0] = sa;
      float cn = ca * c1 - sa * s1, sn = sa * c1 + ca * s1;
      aC[1] = cn; aSf[1] = sn;
      acc = wmma4(aC, bR, acc);
      acc = wmma4(aSf, bS, acc);
      float t = ca * c4 - sa * s4; sa = sa * c4 + ca * s4; ca = t;
    }
    int l = l0 + col;
    #pragma unroll
    for (int r = 0; r < 8; ++r) {
      int n = nt * 16 + r + 8 * half;
      float hann = 0.5f - 0.5f * __cosf((2.f * PI_F / (float)NFFT) * (float)n);
      atomicAdd(&noise[(size_t)b * NOISELEN + (size_t)l * HOP_ + n], acc[r] * hann);
    }
  }
}

// ---------------- finalize: slice + combine into d_out (y, periodic, aperiodic, noise_exc) ----------------
__global__ void k_final(const float* __restrict__ exc, const float* __restrict__ sig,
                        const float* __restrict__ noise, float* __restrict__ out) {
  const size_t N = (size_t)NB * OUTLEN;
  size_t i = (size_t)blockIdx.x * blockDim.x + threadIdx.x;
  size_t s = (size_t)gridDim.x * blockDim.x;
  for (; i < N; i += s) {
    size_t b = i / OUTLEN, t = i - b * OUTLEN;
    float p = sig[b * SIGLEN + t + 120];
    float a = noise[b * NOISELEN + t + 120];
    float e = exc[b * NOISELEN + t + 120];
    out[i]         = p + a;   // y
    out[N + i]     = p;       // periodic
    out[2 * N + i] = a;       // aperiodic
    out[3 * N + i] = e;       // noise_exc
  }
}

extern "C" void kernel_launch(void* const* d_in, const int* in_sizes, int n_in,
                              void* d_out, int out_size, void* d_ws, size_t ws_size,
                              hipStream_t stream) {
  const float* ir    = (const float*)d_in[0];
  const float* ap    = (const float*)d_in[1];
  const float* pitch = (const float*)d_in[2];
  const float* ph0   = (const float*)d_in[3];
  const float* exc   = (const float*)d_in[4];
  float* out = (float*)d_out;

  float* ws = (float*)d_ws;
  float* sig   = ws;
  float* noise = ws + SIG_W;
  int*   cnt   = (int*)(ws + CNT_OFF);
  int*   gT    = (int*)(ws + GT_OFF);
  float* gF    = ws + GF_OFF;

  k_zero <<<2048, 256, 0, stream>>>((unsigned int*)d_ws, (long)ZERO_W);
  k_phase<<<NB, 256, 0, stream>>>(pitch, ph0, cnt, gT, gF);
  k_grain<<<NB * (CAPG / 16), 256, 0, stream>>>(ir, cnt, gT, gF, sig);
  k_noise<<<(NB * LFR) / 16, 256, 0, stream>>>(ap, exc, noise);
  k_final<<<8192, 256, 0, stream>>>(exc, sig, noise, out);
}